// IterativeEdgeModel_32873679684355
// MI455X (gfx1250) — compile-verified
//
#include <hip/hip_runtime.h>
#include <stdint.h>

// ---------------------------------------------------------------------------
// IterativeEdgeModel.assign_clusters for MI455X (gfx1250, wave32).
//
// Segment-argmax scatter: pack (pred_bits, ~edge_idx) into a u64 key and do
// one global_atomic_max_u64 per edge into an L2-resident key array (1.6 MB
// << 192 MB L2).  Nonnegative float bit patterns are monotone as uint32, and
// ~idx in the low half makes the SMALLEST edge index win among equal preds
// under max() -- exactly torch.argmax's first-max tie-break.
//
// Edge stream (pred + dst, 102 MB read once) is staged LDS-side with the
// CDNA5 async copy path (GLOBAL_LOAD_ASYNC_TO_LDS_B128 + s_wait_asynccnt),
// double-buffered per wave, so VMEM-return pressure is removed while the
// VALU drains atomics.  Falls back to plain NT b128 loads if the async
// builtin is not declared by this toolchain.
// Memory-bound floor: ~102 MB / 23.3 TB/s ~= 4.4 us.
// ---------------------------------------------------------------------------

typedef float v4f __attribute__((ext_vector_type(4)));
typedef int   v4i __attribute__((ext_vector_type(4)));

#define THRESH 0.5f
#define TPB  256        // 8 wave32 waves per block
#define WAVE 32
#define WPB  (TPB / WAVE)

#if defined(__has_builtin)
#if __has_builtin(__builtin_amdgcn_global_load_async_to_lds_b128)
#define HAVE_ASYNC_LDS 1
#endif
#endif

__device__ __forceinline__ unsigned long long make_key(float p, unsigned idx) {
  // pred is uniform[0,1) -> nonnegative -> bit pattern order == float order.
  return ((unsigned long long)__float_as_uint(p) << 32) |
         (unsigned long long)(~idx);    // ~idx: smaller idx wins ties on max
}

__global__ void iem_init(unsigned long long* __restrict__ keys,
                         int* __restrict__ flag, int n) {
  int i = blockIdx.x * blockDim.x + threadIdx.x;
  if (i < n) keys[i] = 0ull;            // 0 == "no incoming edge" sentinel
  if (i == 0) *flag = 0;
}

#ifdef HAVE_ASYNC_LDS
// ---- CDNA5 async global->LDS staging path (ASYNCcnt) ----------------------
// Builtin signature (from hipcc diagnostics): param0 = AS(1) v4i* (global
// source), param1 = AS(3) v4i* (LDS dest), then (int offset, int cpol).
typedef __attribute__((address_space(1))) v4i* glb_v4i_p;
typedef __attribute__((address_space(3))) v4i* lds_v4i_p;

// Generic LDS pointers carry the LDS byte offset in their low 32 bits, so an
// integer round-trip yields the 32-bit AS(3) pointer the builtin wants.
static __device__ __forceinline__ lds_v4i_p to_lds(const void* p) {
  return (lds_v4i_p)(uint32_t)(uintptr_t)p;
}
static __device__ __forceinline__ glb_v4i_p to_glb(const void* p) {
  return (glb_v4i_p)(uintptr_t)p;
}

#if __has_builtin(__builtin_amdgcn_s_wait_asynccnt)
#define WAIT_ASYNC(n) __builtin_amdgcn_s_wait_asynccnt(n)
#else
#define WAIT_ASYNC(n) asm volatile("s_wait_asynccnt %0" ::"i"(n) : "memory")
#endif

#define ITERS 8
#define CHUNK (ITERS * WAVE * 4)   // 1024 edges per wave

// Each wave owns a private double-buffered LDS slice: ASYNCcnt is per-wave
// and loads complete in order, so no workgroup barrier is needed.
__global__ void iem_edge_scan_async(const float* __restrict__ pred,
                                    const int* __restrict__ dst,
                                    unsigned long long* __restrict__ keys,
                                    int nchunks) {
  __shared__ __align__(16) char smem[WPB][2][2][WAVE * 16];  // 16 KB / block
  const int wave = threadIdx.x / WAVE;
  const int lane = threadIdx.x % WAVE;
  const int gw   = blockIdx.x * WPB + wave;
  if (gw >= nchunks) return;                 // wave-uniform exit
  const int base = gw * CHUNK;

  auto issue = [&](int t, int b) {
    const int off = base + t * (WAVE * 4) + lane * 4;   // 16B per lane
    __builtin_amdgcn_global_load_async_to_lds_b128(
        to_glb(pred + off), to_lds(&smem[wave][b][0][lane * 16]), 0, 0);
    __builtin_amdgcn_global_load_async_to_lds_b128(
        to_glb(dst + off),  to_lds(&smem[wave][b][1][lane * 16]), 0, 0);
  };

  issue(0, 0);
  for (int t = 0; t < ITERS; ++t) {
    const int b = t & 1;
    if (t + 1 < ITERS) {
      issue(t + 1, b ^ 1);
      WAIT_ASYNC(2);     // in-order completion: first 2 (current tile) done
    } else {
      WAIT_ASYNC(0);
    }
    const v4f p = *(const v4f*)&smem[wave][b][0][lane * 16];  // ds_load_b128
    const v4i d = *(const v4i*)&smem[wave][b][1][lane * 16];
    const unsigned i0 = (unsigned)(base + t * (WAVE * 4) + lane * 4);
    atomicMax(&keys[d.x], make_key(p.x, i0 + 0));
    atomicMax(&keys[d.y], make_key(p.y, i0 + 1));
    atomicMax(&keys[d.z], make_key(p.z, i0 + 2));
    atomicMax(&keys[d.w], make_key(p.w, i0 + 3));
  }
}
#endif  // HAVE_ASYNC_LDS

// ---- fallback: 4 edges/thread via 128-bit non-temporal loads --------------
__global__ void iem_edge_scan_v4(const float* __restrict__ pred,
                                 const int* __restrict__ dst,
                                 unsigned long long* __restrict__ keys,
                                 int e) {
  int i = (blockIdx.x * blockDim.x + threadIdx.x) * 4;
  if (i + 3 < e) {
    v4f p = __builtin_nontemporal_load((const v4f*)(pred + i));
    v4i d = __builtin_nontemporal_load((const v4i*)(dst + i));
    atomicMax(&keys[d.x], make_key(p.x, (unsigned)(i + 0)));
    atomicMax(&keys[d.y], make_key(p.y, (unsigned)(i + 1)));
    atomicMax(&keys[d.z], make_key(p.z, (unsigned)(i + 2)));
    atomicMax(&keys[d.w], make_key(p.w, (unsigned)(i + 3)));
  } else {
    for (; i < e; ++i) {
      float p = __builtin_nontemporal_load(pred + i);
      int   d = __builtin_nontemporal_load(dst + i);
      atomicMax(&keys[d], make_key(p, (unsigned)i));
    }
  }
}

// Scalar kernel for [start, e) tails / unaligned cases.
__global__ void iem_edge_scan_s(const float* __restrict__ pred,
                                const int* __restrict__ dst,
                                unsigned long long* __restrict__ keys,
                                int start, int e) {
  int i = start + blockIdx.x * blockDim.x + threadIdx.x;
  if (i >= e) return;
  float p = __builtin_nontemporal_load(pred + i);
  int   d = __builtin_nontemporal_load(dst + i);
  atomicMax(&keys[d], make_key(p, (unsigned)i));
}

__global__ void iem_resolve(const unsigned long long* __restrict__ keys,
                            const int* __restrict__ src,
                            const int* __restrict__ matched,
                            float* __restrict__ out_matched,
                            float* __restrict__ out_pred,
                            int* __restrict__ flag, int n) {
  int i = blockIdx.x * blockDim.x + threadIdx.x;
  if (i >= n) return;
  unsigned long long key = keys[i];
  bool has_edge = (key != 0ull);
  // Decode: high 32 = pred bits, low 32 = ~edge_idx.
  float win_pred = has_edge ? __uint_as_float((unsigned)(key >> 32)) : 0.0f;
  int m = matched[i];
  int nm = m;
  bool update = (m == -1) & has_edge & (win_pred > THRESH);
  if (update) {
    unsigned win = ~(unsigned)key;      // winning edge index
    nm = matched[src[win]];             // src is in the assigned partition
  }
  // One flag atomic per wave instead of per lane (wave32 ballot).
  unsigned long long bal = __ballot(update);
  if (bal != 0ull &&
      (int)(threadIdx.x & (WAVE - 1)) == (__ffsll((unsigned long long)bal) - 1))
    atomicOr(flag, 1);
  out_matched[i] = (float)nm;           // values < 2^24, exact in f32
  out_pred[i]    = win_pred;
}

__global__ void iem_write_flag(const int* __restrict__ flag,
                               float* __restrict__ out_found) {
  *out_found = (*flag != 0) ? 1.0f : 0.0f;
}

extern "C" void kernel_launch(void* const* d_in, const int* in_sizes, int n_in,
                              void* d_out, int out_size, void* d_ws, size_t ws_size,
                              hipStream_t stream) {
  const float* edge_pred  = (const float*)d_in[0];
  const int*   edge_index = (const int*)d_in[1];   // [2, E] row-major
  const int*   matched    = (const int*)d_in[2];
  const int e = in_sizes[0];
  const int n = in_sizes[2];

  const int* src = edge_index;        // edge_index[0, :]
  const int* dst = edge_index + e;    // edge_index[1, :]

  unsigned long long* keys = (unsigned long long*)d_ws;  // n * 8 bytes
  int* flag = (int*)(keys + n);

  float* out_matched = (float*)d_out;       // [n]
  float* out_pred    = out_matched + n;     // [n]
  float* out_found   = out_pred + n;        // [1]

  iem_init<<<(n + TPB - 1) / TPB, TPB, 0, stream>>>(keys, flag, n);

  const bool al16 = (((uintptr_t)edge_pred & 15) == 0) &&
                    (((uintptr_t)dst & 15) == 0);
  int done = 0;
#ifdef HAVE_ASYNC_LDS
  if (al16) {
    const int nchunks = e / CHUNK;
    if (nchunks > 0) {
      iem_edge_scan_async<<<(nchunks + WPB - 1) / WPB, TPB, 0, stream>>>(
          edge_pred, dst, keys, nchunks);
      done = nchunks * CHUNK;
    }
  }
#endif
  if (done == 0 && al16 && (e & 3) == 0) {
    iem_edge_scan_v4<<<(e / 4 + TPB - 1) / TPB, TPB, 0, stream>>>(
        edge_pred, dst, keys, e);
    done = e;
  }
  if (e > done) {
    iem_edge_scan_s<<<(e - done + TPB - 1) / TPB, TPB, 0, stream>>>(
        edge_pred, dst, keys, done, e);
  }

  iem_resolve<<<(n + TPB - 1) / TPB, TPB, 0, stream>>>(
      keys, src, matched, out_matched, out_pred, flag, n);

  iem_write_flag<<<1, 1, 0, stream>>>(flag, out_found);
}